// Attention_67147518705921
// MI455X (gfx1250) — compile-verified
//
#include <hip/hip_runtime.h>
#include <math.h>

#define BB 32
#define SS 4096
#define HH 1024
#define SC 32                    // sequence rows per chunk
#define NCH (SS / SC)            // 128 chunks per batch
#define TPB 512
#define PAD 4
#define LROW (HH + PAD)          // padded LDS row stride (banks spread)

typedef __attribute__((ext_vector_type(2))) float v2f;
typedef __attribute__((ext_vector_type(8))) float v8f;

// ---------------- primary path: single streaming pass (flash-style) ----------------

__global__ __launch_bounds__(TPB) void attn_chunk_kernel(
    const float* __restrict__ x, const float* __restrict__ w,
    float* __restrict__ cpart, float* __restrict__ mlpart) {
  __shared__ float tile[SC * LROW];     // 32 x 1024 fp32 tile (+pad) ~128.5 KB
  __shared__ float wlds[HH];
  __shared__ float epart[8 * SC];       // per-K-slice partial energies
  __shared__ float parr[SC];

  const int blk = blockIdx.x;
  const int b   = blk / NCH;
  const int ch  = blk % NCH;
  const int s0  = ch * SC;
  const int t   = threadIdx.x;
  const int wv  = t >> 5;
  const int lane = t & 31;

  // ---- Phase A: async DMA global -> LDS (ASYNCcnt path, bypasses VGPRs) ----
  {
    const float* srcf = x + ((long)b * SS + s0) * HH;
    #pragma unroll
    for (int i = 0; i < (SC * HH / 4) / TPB; ++i) {   // 16 x 16B per thread
      int idx = t + i * TPB;
      int row = idx >> 8;          // / (HH/4)
      int c4  = idx & 255;
      // LDS byte offset = low 32 bits of the shared pointer's flat address
      unsigned lds_off = (unsigned)(uintptr_t)(&tile[row * LROW + c4 * 4]);
      unsigned long long gaddr =
          (unsigned long long)(uintptr_t)(srcf + row * HH + c4 * 4);
      asm volatile("global_load_async_to_lds_b128 %0, %1, off"
                   :: "v"(lds_off), "v"(gaddr) : "memory");
    }
    wlds[t]       = w[t];
    wlds[t + TPB] = w[t + TPB];
    asm volatile("s_wait_asynccnt 0x0" ::: "memory");
  }
  __syncthreads();

  // ---- Phase 1: energies e[s] = dot(x[s,:], w) via V_WMMA_F32_16X16X4_F32 ----
  // 16 waves = 2 row-tiles x 8 K-slices (128 K each). w is broadcast across N,
  // so every column of D holds the same dot -> trivial extraction.
  {
    const int tile16 = wv & 1;
    const int slice  = wv >> 1;
    const int rowA   = tile16 * 16 + (lane & 15);
    const int koff   = (lane < 16) ? 0 : 2;          // ISA A layout: K01 | K23
    const float* arow = &tile[rowA * LROW];
    v8f c = {0.f, 0.f, 0.f, 0.f, 0.f, 0.f, 0.f, 0.f};
    int kb = slice * 128;
    #pragma unroll 4
    for (int i = 0; i < 32; ++i, kb += 4) {
      float2 a2 = *(const float2*)&arow[kb + koff];
      float2 b2 = *(const float2*)&wlds[kb + koff];
      v2f av; av.x = a2.x; av.y = a2.y;
      v2f bv; bv.x = b2.x; bv.y = b2.y;
      c = __builtin_amdgcn_wmma_f32_16x16x4_f32(false, av, false, bv,
                                                (short)0, c, false, false);
    }
    // D layout: lanes 0-15 hold M=v in VGPR v; lanes 16-31 hold M=8+v.
    if ((lane & 15) == 0) {
      int mb = tile16 * 16 + ((lane >> 4) ? 8 : 0);
      #pragma unroll
      for (int v = 0; v < 8; ++v) epart[slice * SC + mb + v] = c[v];
    }
  }
  __syncthreads();

  // ---- Phase 2: tanh + chunk-local softmax stats (wave 0, one lane per row) ----
  if (wv == 0) {
    float e = 0.f;
    #pragma unroll
    for (int j = 0; j < 8; ++j) e += epart[j * SC + lane];
    e = tanhf(e);
    float m = e;
    for (int off = 16; off >= 1; off >>= 1) m = fmaxf(m, __shfl_xor(m, off, 32));
    float p = __expf(e - m);
    float l = p;
    for (int off = 16; off >= 1; off >>= 1) l += __shfl_xor(l, off, 32);
    parr[lane] = p;
    if (lane == 0) {
      mlpart[((long)b * NCH + ch) * 2 + 0] = m;
      mlpart[((long)b * NCH + ch) * 2 + 1] = l;
    }
  }
  __syncthreads();

  // ---- Phase 3: partial context from LDS (tile never re-read from HBM) ----
  {
    int h0 = t * 2;
    float c0 = 0.f, c1 = 0.f;
    #pragma unroll 8
    for (int s = 0; s < SC; ++s) {
      float p = parr[s];
      float2 xv = *(const float2*)&tile[s * LROW + h0];
      c0 += p * xv.x;
      c1 += p * xv.y;
    }
    float2 o; o.x = c0; o.y = c1;
    *(float2*)&cpart[((long)b * NCH + ch) * HH + h0] = o;
  }
}

__global__ __launch_bounds__(TPB) void attn_combine_kernel(
    const float* __restrict__ cpart, const float* __restrict__ mlpart,
    float* __restrict__ out) {
  __shared__ float sm[NCH];
  __shared__ float sl[NCH];
  __shared__ float scale[NCH];
  __shared__ float sinvL;
  const int b = blockIdx.x;
  const int t = threadIdx.x;
  if (t < NCH) {
    sm[t] = mlpart[((long)b * NCH + t) * 2 + 0];
    sl[t] = mlpart[((long)b * NCH + t) * 2 + 1];
  }
  __syncthreads();
  if (t == 0) {
    float M = -1e30f;
    for (int i = 0; i < NCH; ++i) M = fmaxf(M, sm[i]);
    float L = 0.f;
    for (int i = 0; i < NCH; ++i) {
      float sc = __expf(sm[i] - M);
      scale[i] = sc;
      L += sc * sl[i];
    }
    sinvL = 1.f / L;
  }
  __syncthreads();
  int h0 = t * 2;
  float a0 = 0.f, a1 = 0.f;
  for (int chn = 0; chn < NCH; ++chn) {
    float sc = scale[chn];
    float2 v = *(const float2*)&cpart[((long)b * NCH + chn) * HH + h0];
    a0 += sc * v.x;
    a1 += sc * v.y;
  }
  float invL = sinvL;
  float2 o; o.x = a0 * invL; o.y = a1 * invL;
  *(float2*)&out[(long)b * HH + h0] = o;
}

// ---------------- fallback path (if d_ws < ~16 MB): classic two-pass ----------------

__global__ __launch_bounds__(256) void fb_energy(const float* __restrict__ x,
                                                 const float* __restrict__ w,
                                                 float* __restrict__ e) {
  const int wv = threadIdx.x >> 5;
  const int lane = threadIdx.x & 31;
  const long row = (long)blockIdx.x * 8 + wv;
  const float4* xp = (const float4*)(x + row * HH);
  const float4* wp = (const float4*)w;
  float acc = 0.f;
  for (int i = lane; i < HH / 4; i += 32) {
    float4 v = xp[i];
    float4 wv4 = wp[i];
    acc += v.x * wv4.x + v.y * wv4.y + v.z * wv4.z + v.w * wv4.w;
  }
  for (int off = 16; off >= 1; off >>= 1) acc += __shfl_xor(acc, off, 32);
  if (lane == 0) e[row] = tanhf(acc);
}

__global__ __launch_bounds__(256) void fb_softmax(float* __restrict__ e) {
  __shared__ float red[256];
  const int b = blockIdx.x, t = threadIdx.x;
  float* eb = e + (long)b * SS;
  float m = -1e30f;
  for (int s = t; s < SS; s += 256) m = fmaxf(m, eb[s]);
  red[t] = m; __syncthreads();
  for (int off = 128; off >= 1; off >>= 1) {
    if (t < off) red[t] = fmaxf(red[t], red[t + off]);
    __syncthreads();
  }
  float M = red[0]; __syncthreads();
  float l = 0.f;
  for (int s = t; s < SS; s += 256) l += __expf(eb[s] - M);
  red[t] = l; __syncthreads();
  for (int off = 128; off >= 1; off >>= 1) {
    if (t < off) red[t] += red[t + off];
    __syncthreads();
  }
  float invL = 1.f / red[0];
  __syncthreads();
  for (int s = t; s < SS; s += 256) eb[s] = __expf(eb[s] - M) * invL;
}

__global__ __launch_bounds__(1024) void fb_context(const float* __restrict__ x,
                                                   const float* __restrict__ p,
                                                   float* __restrict__ out) {
  const int b = blockIdx.x;
  const int h = threadIdx.x;
  const float* xb = x + (long)b * SS * HH;
  const float* pb = p + (long)b * SS;
  float acc = 0.f;
  for (int s = 0; s < SS; ++s) acc += pb[s] * xb[(long)s * HH + h];
  out[(long)b * HH + h] = acc;
}

// ---------------- host launcher ----------------

extern "C" void kernel_launch(void* const* d_in, const int* in_sizes, int n_in,
                              void* d_out, int out_size, void* d_ws, size_t ws_size,
                              hipStream_t stream) {
  (void)in_sizes; (void)n_in; (void)out_size;
  const float* x = (const float*)d_in[0];
  const float* w = (const float*)d_in[1];
  float* out = (float*)d_out;

  const size_t need = ((size_t)BB * NCH * HH + (size_t)BB * NCH * 2) * sizeof(float);
  if (ws_size >= need) {
    float* cpart  = (float*)d_ws;
    float* mlpart = cpart + (size_t)BB * NCH * HH;
    attn_chunk_kernel<<<BB * NCH, TPB, 0, stream>>>(x, w, cpart, mlpart);
    attn_combine_kernel<<<BB, TPB, 0, stream>>>(cpart, mlpart, out);
  } else {
    float* e = (float*)d_ws;  // needs B*S*4 = 512 KB
    fb_energy<<<(BB * SS) / 8, 256, 0, stream>>>(x, w, e);
    fb_softmax<<<BB, 256, 0, stream>>>(e);
    fb_context<<<BB, HH, 0, stream>>>(x, e, out);
  }
}